// Attn_1176821040084
// MI455X (gfx1250) — compile-verified
//
#include <hip/hip_runtime.h>

typedef __attribute__((ext_vector_type(2))) float v2f;
typedef __attribute__((ext_vector_type(8))) float v8f;

#define DDIM 2048
#define SDIM 16384

// ---------------- Phase 1a: partial u = v^T W over 256-row chunks ----------------
// grid (DDIM/256, 8), block 256.  part[chunk][d] = sum_{e in chunk} v[e]*W[e,d]
__global__ __launch_bounds__(256) void vtw_partial_kernel(
    const float* __restrict__ W, const float* __restrict__ v,
    float* __restrict__ part) {
  const int d  = blockIdx.x * 256 + threadIdx.x;
  const int e0 = blockIdx.y * 256;
  float acc = 0.f;
#pragma unroll 4
  for (int e = 0; e < 256; ++e) {
    acc = fmaf(v[e0 + e], W[(size_t)(e0 + e) * DDIM + d], acc);
  }
  part[blockIdx.y * DDIM + d] = acc;
}

// ---------------- Phase 1b: reduce 8 partials -> u[d] ----------------
__global__ __launch_bounds__(256) void u_reduce_kernel(
    const float* __restrict__ part, float* __restrict__ u) {
  const int d = blockIdx.x * 256 + threadIdx.x;
  float acc = 0.f;
#pragma unroll
  for (int j = 0; j < 8; ++j) acc += part[j * DDIM + d];
  u[d] = acc;
}

// ---------------- Phase 2: z = outputs @ u via V_WMMA_F32_16X16X4_F32 ----------------
// Broadcast-B trick: B's 16 columns all hold u[k0..k0+3], so every column of the
// 16x16 accumulator equals the exact fp32 dot product for its row.
// Wave w of block b owns rows s0 = (b*8+w)*16 .. +15.
__global__ __launch_bounds__(256) void matvec_wmma_kernel(
    const float* __restrict__ outputs, const float* __restrict__ u,
    float* __restrict__ z) {
  __shared__ float lds_u[DDIM];
  const int tid = threadIdx.x;
  for (int i = tid; i < DDIM; i += 256) lds_u[i] = u[i];
  __syncthreads();

  const int wave = tid >> 5;
  const int lane = tid & 31;
  const int m    = lane & 15;          // row within tile (A layout: lane = M)
  const int koff = (lane >> 4) << 1;   // lanes 0-15 -> K 0,1 ; lanes 16-31 -> K 2,3
  const int s0   = (blockIdx.x * 8 + wave) * 16;

  const float* Arow = outputs + (size_t)(s0 + m) * DDIM + koff;

  v8f c = {};
#pragma unroll 8
  for (int k0 = 0; k0 < DDIM; k0 += 4) {
    v2f a  = *(const v2f*)(Arow + k0);
    v2f bb = *(const v2f*)(&lds_u[k0 + koff]);
    // (neg_a, A, neg_b, B, c_mod, C, reuse_a, reuse_b)
    c = __builtin_amdgcn_wmma_f32_16x16x4_f32(false, a, false, bb,
                                              (short)0, c, false, false);
  }

  // C/D layout: VGPR r -> (lane 0..15: M=r, N=lane) (lane 16..31: M=8+r).
  // All columns identical; read column N=0 from lane 0 and lane 16.
  if (lane == 0) {
#pragma unroll
    for (int r = 0; r < 8; ++r) z[s0 + r] = c[r];
  } else if (lane == 16) {
#pragma unroll
    for (int r = 0; r < 8; ++r) z[s0 + 8 + r] = c[r];
  }
}

// ---------------- Phase 3: softmax over SDIM values, single block ----------------
__global__ __launch_bounds__(1024) void softmax_kernel(
    const float* __restrict__ z, float* __restrict__ out) {
  __shared__ float red[1024];
  const int tid = threadIdx.x;

  float vals[16];
  float lmax = -3.402823466e+38f;
#pragma unroll
  for (int i = 0; i < 16; ++i) {
    vals[i] = z[i * 1024 + tid];
    lmax = fmaxf(lmax, vals[i]);
  }
  red[tid] = lmax;
  __syncthreads();
  for (int s = 512; s > 0; s >>= 1) {
    if (tid < s) red[tid] = fmaxf(red[tid], red[tid + s]);
    __syncthreads();
  }
  const float gmax = red[0];
  __syncthreads();

  float lsum = 0.f;
#pragma unroll
  for (int i = 0; i < 16; ++i) {
    vals[i] = __expf(vals[i] - gmax);
    lsum += vals[i];
  }
  red[tid] = lsum;
  __syncthreads();
  for (int s = 512; s > 0; s >>= 1) {
    if (tid < s) red[tid] += red[tid + s];
    __syncthreads();
  }
  const float inv = 1.0f / red[0];

#pragma unroll
  for (int i = 0; i < 16; ++i) out[i * 1024 + tid] = vals[i] * inv;
}

extern "C" void kernel_launch(void* const* d_in, const int* in_sizes, int n_in,
                              void* d_out, int out_size, void* d_ws, size_t ws_size,
                              hipStream_t stream) {
  const float* outputs = (const float*)d_in[0];  // [S, D]
  const float* W       = (const float*)d_in[1];  // [D, D]
  // d_in[2] is the bias b: softmax is shift-invariant, so dot(b,v) cancels.
  const float* wvec    = (const float*)d_in[3];  // [D]
  float* out = (float*)d_out;                    // [S]

  float* ws   = (float*)d_ws;
  float* part = ws;                 // 8 * DDIM floats
  float* u    = ws + 8 * DDIM;      // DDIM floats
  float* z    = ws + 9 * DDIM;      // SDIM floats

  vtw_partial_kernel<<<dim3(DDIM / 256, 8), 256, 0, stream>>>(W, wvec, part);
  u_reduce_kernel<<<DDIM / 256, 256, 0, stream>>>(part, u);
  matvec_wmma_kernel<<<SDIM / 128, 256, 0, stream>>>(outputs, u, z);
  softmax_kernel<<<1, 1024, 0, stream>>>(z, out);
}